// ResidueBasedGroup_31928786878756
// MI455X (gfx1250) — compile-verified
//
#include <hip/hip_runtime.h>

// Problem constants fixed by the reference (setup_inputs):
//   B=32, N=32768, R=4096, ATOM_DIMS=11, MAX_ATOMS=24
// Outputs concatenated flat in return order:
//   neighborhood (B,R,24,11) | centers (B,R,3) | residue_mask (B,R)
#define B_  32
#define N_  32768
#define R_  4096
#define D_  11
#define MA_ 24

static_assert((R_ & (R_ - 1)) == 0, "R must be a power of two for shift indexing");

__global__ __launch_bounds__(256) void residue_group_kernel(
    const float* __restrict__ points,        // (B, N, 11)
    const long long* __restrict__ starts,    // (B, R) int64, sorted per batch
    float* __restrict__ out)
{
    const int lane = threadIdx.x & 31;       // wave32 (gfx1250)
    const int wave = threadIdx.x >> 5;
    const long long rid = (long long)blockIdx.x * 8 + wave;   // residue id in [0, B*R)
    const int b = (int)(rid >> 12);          // / R_ (4096)
    const int r = (int)(rid & (R_ - 1));     // % R_

    const long long sidx = (long long)b * R_ + r;
    const int start = (int)starts[sidx];
    const int end   = (r == R_ - 1) ? N_ : (int)starts[sidx + 1];
    const int count = end - start;           // >= 0 (starts are sorted)

    const float* __restrict__ bp = points + ((long long)b * N_ + start) * D_;

    // ---- Phase 1: segment sum of xyz over the FULL segment [start, end) ----
    float sx = 0.0f, sy = 0.0f, sz = 0.0f;
    for (int i = lane; i < count; i += 32) {
        const float* p = bp + (long long)i * D_;
        // CDNA5 global_prefetch_b8: stay a stride ahead of the streaming loads
        __builtin_prefetch(p + 32 * D_, 0, 0);
        sx += p[0];
        sy += p[1];
        sz += p[2];
    }
    // wave32 butterfly reduction -> every lane holds the full sum
    #pragma unroll
    for (int off = 16; off > 0; off >>= 1) {
        sx += __shfl_xor(sx, off, 32);
        sy += __shfl_xor(sy, off, 32);
        sz += __shfl_xor(sz, off, 32);
    }

    const float inv = (count > 0) ? (1.0f / (float)count) : 0.0f;
    const float cx = sx * inv;
    const float cy = sy * inv;
    const float cz = sz * inv;

    // ---- Phase 2: centers + mask (lane 0 only; tiny traffic) ----
    const long long NB = (long long)B_ * R_ * MA_ * D_;        // centers offset
    const long long CB = NB + (long long)B_ * R_ * 3;          // mask offset
    if (lane == 0) {
        float* c = out + NB + sidx * 3;
        c[0] = cx;
        c[1] = cy;
        c[2] = cz;
        out[CB + sidx] = (count > 0) ? 1.0f : 0.0f;
    }

    // ---- Phase 3: neighborhood gather: 24 atoms x 11 dims = 264 floats ----
    // Re-reads hit L2 (points tensor is 46 MB, L2 is 192 MB). Stores are the
    // dominant traffic and are fully coalesced (lanes write consecutive dwords).
    float* __restrict__ nb = out + sidx * (MA_ * D_);
    for (int t = lane; t < MA_ * D_; t += 32) {
        const int k = t / D_;                // constant div -> mul/shift
        const int d = t - k * D_;
        float v = 0.0f;
        if (k < count) {                     // k < count <= N-start, always in-bounds
            v = bp[(long long)k * D_ + d];
            if (d == 0)      v -= cx;
            else if (d == 1) v -= cy;
            else if (d == 2) v -= cz;
        }
        nb[t] = v;
    }
}

extern "C" void kernel_launch(void* const* d_in, const int* in_sizes, int n_in,
                              void* d_out, int out_size, void* d_ws, size_t ws_size,
                              hipStream_t stream)
{
    const float*     points = (const float*)d_in[0];      // float32 (B,N,11)
    const long long* starts = (const long long*)d_in[1];  // int64   (B,R)
    float*           out    = (float*)d_out;              // float32, concat outputs

    (void)in_sizes; (void)n_in; (void)out_size; (void)d_ws; (void)ws_size;

    // 1 wave per residue, 8 waves (256 threads) per block
    const int n_res  = B_ * R_;              // 131072
    const int blocks = n_res / 8;            // 16384
    residue_group_kernel<<<blocks, 256, 0, stream>>>(points, starts, out);
}